// DemoProj_30605936951454
// MI455X (gfx1250) — compile-verified
//
#include <hip/hip_runtime.h>

// out[b, i, f] = x[b, i] * W[i, f] + b[i, f]
// BS=16384, DEMO_DIM=32, FEAT_DIM=256  -> 512 MB f32 output, store-BW bound.
// Floor ~= 512MB / 23.3 TB/s ~= 22 us; compute is free -> feed it to the WMMA pipe.

#define BS       16384
#define DEMO_DIM 32
#define FEAT_DIM 256
#define ROW      (DEMO_DIM * FEAT_DIM)   // 8192 floats per batch row

typedef __attribute__((ext_vector_type(2))) float v2f;
typedef __attribute__((ext_vector_type(8))) float v8f;

// One wave per (16-batch-tile, i) pair, looping over 16 feature tiles of 16.
// Each iteration is one V_WMMA_F32_16X16X4_F32 using TWO live K slots:
//   A[m,0] = x[btile*16+m, i]     A[m,1] = 1.0      (K=2,3 zero)
//   B[0,n] = W[i, ft*16+n]        B[1,n] = bias[i, ft*16+n]
//   C      = 0 (inline constant -> no accumulator init movs)
// D[m,n] = x*W + 1*bias  == the fused broadcast multiply-add.
//
// 16x4 f32 A layout: VGPR0 lanes0-15=K0 / lanes16-31=K2; VGPR1 K1/K3.
// B symmetric per-lane(N). Lanes 16-31 (K=2,3) are zeroed with v_cndmask
// selects — no exec divergence, EXEC stays all-1s (WMMA requirement).
__global__ __launch_bounds__(256) void demo_proj_wmma_kernel(
    const float* __restrict__ x,
    const float* __restrict__ W,
    const float* __restrict__ bias,
    float* __restrict__ out)
{
    const int lane   = threadIdx.x & 31;
    const int wv     = threadIdx.x >> 5;                 // wave in block (0..7)
    const int gwave  = blockIdx.x * 8 + wv;              // 0 .. 32767
    const int i      = gwave & (DEMO_DIM - 1);           // demo index 0..31
    const int btile  = gwave >> 5;                       // 0..1023 (16 rows each)
    const bool hi    = (lane & 16) != 0;                 // lanes 16-31 (K=2,3)
    const int n      = lane & 15;                        // tile column / row

    // A: K=0 column = x values, K=1 column = 1.0 (bias multiplier).
    // Load for ALL lanes (lanes 16-31 duplicate lanes 0-15: valid, cached),
    // then select-to-zero -> v_cndmask, no branches.
    const float xv = x[(btile * 16 + n) * DEMO_DIM + i];
    v2f a;
    a.x = hi ? 0.0f : xv;      // VGPR0: K=0 (x) / K=2 (0)
    a.y = hi ? 0.0f : 1.0f;    // VGPR1: K=1 (1.0) / K=3 (0)

    const float* Wrow = W    + i * FEAT_DIM;
    const float* Brow = bias + i * FEAT_DIM;
    float* outbase = out + (size_t)btile * 16 * ROW + (size_t)i * FEAT_DIM;

#pragma unroll 4
    for (int ft = 0; ft < FEAT_DIM / 16; ++ft) {
        const int f = ft * 16 + n;

        // B: K=0 row = W, K=1 row = bias; K=2,3 zero.
        const float wval = Wrow[f];          // unconditional, all lanes
        const float bv   = Brow[f];
        v2f bm;
        bm.x = hi ? 0.0f : wval;             // VGPR0: K=0 / K=2
        bm.y = hi ? 0.0f : bv;               // VGPR1: K=1 / K=3

        // C = 0 -> inline constant SRC2, no accumulator broadcast movs.
        v8f c = {};

        // 8 args: (neg_a, A, neg_b, B, c_mod, C, reuse_a, reuse_b)
        v8f d = __builtin_amdgcn_wmma_f32_16x16x4_f32(
            false, a, false, bm, (short)0, c, false, false);

        // D layout: VGPR v, lane l -> M = v + 8*(l>=16), N = l&15.
        // Non-temporal: 512 MB write-once stream, keep it out of L2.
        const size_t mbase = hi ? (size_t)8 * ROW : 0;
#pragma unroll
        for (int v = 0; v < 8; ++v) {
            __builtin_nontemporal_store(
                d[v], outbase + mbase + (size_t)v * ROW + f);
        }
    }
}

extern "C" void kernel_launch(void* const* d_in, const int* in_sizes, int n_in,
                              void* d_out, int out_size, void* d_ws, size_t ws_size,
                              hipStream_t stream) {
    const float* x    = (const float*)d_in[0];   // [BS, DEMO_DIM]
    const float* W    = (const float*)d_in[1];   // [DEMO_DIM, FEAT_DIM]
    const float* bias = (const float*)d_in[2];   // [DEMO_DIM, FEAT_DIM]
    float* out        = (float*)d_out;           // [BS, DEMO_DIM, FEAT_DIM]

    // 16384/16 batch tiles * 32 i = 32768 waves; 8 waves per 256-thread block.
    const int num_blocks = (BS / 16) * DEMO_DIM / 8;   // 4096
    demo_proj_wmma_kernel<<<num_blocks, 256, 0, stream>>>(x, W, bias, out);
}